// ChebNet_29386166239465
// MI455X (gfx1250) — compile-verified
//
#include <hip/hip_runtime.h>
#include <math.h>

typedef float v2f __attribute__((ext_vector_type(2)));
typedef float v8f __attribute__((ext_vector_type(8)));

#define WMMA_F32X4(a, b, c) \
  __builtin_amdgcn_wmma_f32_16x16x4_f32(false, (a), false, (b), (short)0, (c), false, false)

// ---------------- utility ----------------
__global__ void zero_kernel(float* __restrict__ p, long long n) {
  long long i = (long long)blockIdx.x * blockDim.x + threadIdx.x;
  if (i < n) p[i] = 0.0f;
}

// ---------------- graph norm ----------------
__global__ void deg_kernel(const int* __restrict__ src, float* __restrict__ deg, int E) {
  int e = blockIdx.x * blockDim.x + threadIdx.x;
  if (e < E) unsafeAtomicAdd(&deg[src[e]], 1.0f);
}

__global__ void dinv_kernel(const float* __restrict__ deg, float* __restrict__ dinv, int N) {
  int i = blockIdx.x * blockDim.x + threadIdx.x;
  if (i < N) {
    float d = deg[i];
    dinv[i] = (d > 0.0f) ? rsqrtf(fmaxf(d, 1.0f)) : 0.0f;
  }
}

__global__ void norm_kernel(const int* __restrict__ src, const int* __restrict__ dst,
                            const float* __restrict__ dinv, float* __restrict__ nrm, int E) {
  int e = blockIdx.x * blockDim.x + threadIdx.x;
  if (e < E) nrm[e] = -dinv[src[e]] * dinv[dst[e]];
}

// ---------------- layer 1 GEMM: pre1 = x@W1[0]+b1 ; y1 = x@W1[1] ----------------
// fp32 WMMA 16x16x4. One wave handles a 16-node row tile; D=64 (16 K-steps), H=16.
__global__ void gemm1_kernel(const float* __restrict__ x, const float* __restrict__ W1,
                             const float* __restrict__ b1, float* __restrict__ pre1,
                             float* __restrict__ y1, int N, int nTiles) {
  const int wave = (blockIdx.x * blockDim.x + threadIdx.x) >> 5;
  if (wave >= nTiles) return;                 // wave-uniform exit, EXEC stays all-1
  const int lane = threadIdx.x & 31;
  const int n = lane & 15;                    // column (H) / A-row index
  const int hi = lane >> 4;
  const int base = wave * 16;
  const int mrow = min(base + n, N - 1);      // clamped A row (no divergence)
  const float* __restrict__ W0 = W1;          // [64][16]
  const float* __restrict__ Wb = W1 + 64 * 16;
  const float* __restrict__ xr = x + (long long)mrow * 64;

  v8f acc0 = {0.f, 0.f, 0.f, 0.f, 0.f, 0.f, 0.f, 0.f};
  v8f acc1 = acc0;
#pragma unroll
  for (int kb = 0; kb < 64; kb += 4) {
    const int k0 = kb + 2 * hi;               // per-lane K pair: k0, k0+1
    v2f a = *(const v2f*)(xr + k0);           // A[m][k0], A[m][k0+1]
    v2f b0, bb;
    b0.x = W0[k0 * 16 + n];  b0.y = W0[(k0 + 1) * 16 + n];
    bb.x = Wb[k0 * 16 + n];  bb.y = Wb[(k0 + 1) * 16 + n];
    acc0 = WMMA_F32X4(a, b0, acc0);
    acc1 = WMMA_F32X4(a, bb, acc1);
  }
  const float bias = b1[n];
#pragma unroll
  for (int r = 0; r < 8; ++r) {
    const int row = base + r + 8 * hi;        // C layout: M = r + 8*hi, N = lane&15
    if (row < N) {
      pre1[(long long)row * 16 + n] = acc0[r] + bias;
      y1[(long long)row * 16 + n]  = acc1[r];
    }
  }
}

// ---------------- edge scatter, layer 1 (16 feats/edge) ----------------
__global__ void scatter1_kernel(const int* __restrict__ src, const int* __restrict__ dst,
                                const float* __restrict__ nrm, const float* __restrict__ y1,
                                float* __restrict__ agg1, int E) {
  unsigned tid = blockIdx.x * blockDim.x + threadIdx.x;
  unsigned e = tid >> 4;
  unsigned f = tid & 15u;
  if (e < (unsigned)E) {
    int s = src[e], d = dst[e];
    unsafeAtomicAdd(&agg1[(long long)d * 16 + f], nrm[e] * y1[(long long)s * 16 + f]);
  }
}

// ---------------- layer 2 GEMM (ReLU fused): pre2 = h@W2[0]+b2 ; y2 = h@W2[1] ----------------
// h = relu(pre1 + agg1). K=16 (4 steps), C=40 -> 3 column tiles of 16 (clamped loads).
__global__ void gemm2_kernel(const float* __restrict__ pre1, const float* __restrict__ agg1,
                             const float* __restrict__ W2, const float* __restrict__ b2,
                             float* __restrict__ pre2, float* __restrict__ y2,
                             int N, int nTiles) {
  const int wave = (blockIdx.x * blockDim.x + threadIdx.x) >> 5;
  if (wave >= nTiles) return;
  const int lane = threadIdx.x & 31;
  const int n = lane & 15;
  const int hi = lane >> 4;
  const int base = wave * 16;
  const int mrow = min(base + n, N - 1);
  const float* __restrict__ W0 = W2;          // [16][40]
  const float* __restrict__ Wb = W2 + 16 * 40;

  // A tile: relu(pre1 + agg1), K = 16
  v2f a[4];
#pragma unroll
  for (int i = 0; i < 4; ++i) {
    const int k0 = i * 4 + 2 * hi;
    const long long idx = (long long)mrow * 16 + k0;
    v2f p = *(const v2f*)(pre1 + idx);
    v2f g = *(const v2f*)(agg1 + idx);
    a[i].x = fmaxf(p.x + g.x, 0.0f);
    a[i].y = fmaxf(p.y + g.y, 0.0f);
  }

  v8f acc[6];
#pragma unroll
  for (int i = 0; i < 6; ++i) acc[i] = (v8f){0.f, 0.f, 0.f, 0.f, 0.f, 0.f, 0.f, 0.f};

#pragma unroll
  for (int ct = 0; ct < 3; ++ct) {
    const int col = ct * 16 + n;
    const int cc = min(col, 39);              // clamp: branch-free, garbage cols never stored
#pragma unroll
    for (int i = 0; i < 4; ++i) {
      const int k0 = i * 4 + 2 * hi;
      v2f b0, bb;
      b0.x = W0[k0 * 40 + cc];  b0.y = W0[(k0 + 1) * 40 + cc];
      bb.x = Wb[k0 * 40 + cc];  bb.y = Wb[(k0 + 1) * 40 + cc];
      acc[ct * 2]     = WMMA_F32X4(a[i], b0, acc[ct * 2]);
      acc[ct * 2 + 1] = WMMA_F32X4(a[i], bb, acc[ct * 2 + 1]);
    }
  }

  // all WMMA done; divergent guarded stores are now safe
#pragma unroll
  for (int ct = 0; ct < 3; ++ct) {
    const int col = ct * 16 + n;
    if (col < 40) {
      const float bias = b2[col];
#pragma unroll
      for (int r = 0; r < 8; ++r) {
        const int row = base + r + 8 * hi;
        if (row < N) {
          pre2[(long long)row * 40 + col] = acc[ct * 2][r] + bias;
          y2[(long long)row * 40 + col]   = acc[ct * 2 + 1][r];
        }
      }
    }
  }
}

// ---------------- edge scatter, layer 2 (40 feats/edge) ----------------
__global__ void scatter2_kernel(const int* __restrict__ src, const int* __restrict__ dst,
                                const float* __restrict__ nrm, const float* __restrict__ y2,
                                float* __restrict__ agg2, int E) {
  unsigned tid = blockIdx.x * blockDim.x + threadIdx.x;
  unsigned e = tid / 40u;
  unsigned f = tid - e * 40u;
  if (e < (unsigned)E) {
    int s = src[e], d = dst[e];
    unsafeAtomicAdd(&agg2[(long long)d * 40 + f], nrm[e] * y2[(long long)s * 40 + f]);
  }
}

// ---------------- log_softmax: one wave32 per node over C=40 ----------------
__global__ void lsm_kernel(const float* __restrict__ pre2, const float* __restrict__ agg2,
                           float* __restrict__ out, int N) {
  const int node = (blockIdx.x * blockDim.x + threadIdx.x) >> 5;  // wave-uniform
  if (node >= N) return;
  const int lane = threadIdx.x & 31;
  const float* __restrict__ pr = pre2 + (long long)node * 40;
  const float* __restrict__ ag = agg2 + (long long)node * 40;
  float v0 = pr[lane] + ag[lane];
  float v1 = (lane < 8) ? (pr[lane + 32] + ag[lane + 32]) : -INFINITY;
  float m = fmaxf(v0, v1);
#pragma unroll
  for (int o = 16; o > 0; o >>= 1) m = fmaxf(m, __shfl_xor(m, o, 32));
  float s = expf(v0 - m) + ((lane < 8) ? expf(v1 - m) : 0.0f);
#pragma unroll
  for (int o = 16; o > 0; o >>= 1) s += __shfl_xor(s, o, 32);
  const float lse = m + logf(s);
  float* __restrict__ op = out + (long long)node * 40;
  op[lane] = v0 - lse;
  if (lane < 8) op[lane + 32] = v1 - lse;
}

// ---------------- launcher ----------------
extern "C" void kernel_launch(void* const* d_in, const int* in_sizes, int n_in,
                              void* d_out, int out_size, void* d_ws, size_t ws_size,
                              hipStream_t stream) {
  const float* x  = (const float*)d_in[0];
  const int* ei   = (const int*)d_in[1];
  const float* W1 = (const float*)d_in[2];
  const float* b1 = (const float*)d_in[3];
  const float* W2 = (const float*)d_in[4];
  const float* b2 = (const float*)d_in[5];
  float* out = (float*)d_out;

  const int N = in_sizes[0] / 64;   // 50000
  const int E = in_sizes[1] / 2;    // 800000
  const int* src = ei;
  const int* dst = ei + E;

  // workspace layout (floats)
  float* ws = (float*)d_ws;
  size_t off = 0;
  float* deg  = ws + off; off += (size_t)N;
  float* dinv = ws + off; off += (size_t)N;
  float* nrm  = ws + off; off += (size_t)E;
  float* pre1 = ws + off; off += (size_t)N * 16;
  float* y1   = ws + off; off += (size_t)N * 16;
  float* agg1 = ws + off; off += (size_t)N * 16;
  float* pre2 = ws + off; off += (size_t)N * 40;
  float* y2   = ws + off; off += (size_t)N * 40;
  float* agg2 = ws + off; off += (size_t)N * 40;

  const int B = 256;
  auto blk = [](long long n, int b) { return (unsigned)((n + b - 1) / b); };

  // zero accumulators (must be redone every call)
  zero_kernel<<<blk(N, B), B, 0, stream>>>(deg, N);
  zero_kernel<<<blk((long long)N * 16, B), B, 0, stream>>>(agg1, (long long)N * 16);
  zero_kernel<<<blk((long long)N * 40, B), B, 0, stream>>>(agg2, (long long)N * 40);

  // symmetric-normalization coefficients
  deg_kernel<<<blk(E, B), B, 0, stream>>>(src, deg, E);
  dinv_kernel<<<blk(N, B), B, 0, stream>>>(deg, dinv, N);
  norm_kernel<<<blk(E, B), B, 0, stream>>>(src, dst, dinv, nrm, E);

  // layer 1
  const int tiles = (N + 15) / 16;
  gemm1_kernel<<<blk((long long)tiles * 32, B), B, 0, stream>>>(x, W1, b1, pre1, y1, N, tiles);
  scatter1_kernel<<<blk((long long)E * 16, B), B, 0, stream>>>(src, dst, nrm, y1, agg1, E);

  // layer 2 (ReLU fused into GEMM2 A-load)
  gemm2_kernel<<<blk((long long)tiles * 32, B), B, 0, stream>>>(pre1, agg1, W2, b2, pre2, y2, N, tiles);
  scatter2_kernel<<<blk((long long)E * 40, B), B, 0, stream>>>(src, dst, nrm, y2, agg2, E);

  // log_softmax
  lsm_kernel<<<blk((long long)N * 32, B), B, 0, stream>>>(pre2, agg2, out, N);
}